// QuadraticAttentionModule_65515431133908
// MI455X (gfx1250) — compile-verified
//
#include <hip/hip_runtime.h>

// Problem constants (from reference): B=64, S=32, D=32
#define BATCH    64
#define IN_DIM   1024              // S*D
#define TRIU     524800            // (IN_DIM + IN_DIM^2)/2
#define WSTRIDE  525824            // IN_DIM + TRIU  (K total, W row stride)
#define OUT_DIM  1024              // 32*S
#define XS       1028              // padded LDS row stride: 1028 % 64 == 4 ->
                                   // bank = (4*b + c) % 64, conflict-free reads

typedef __attribute__((ext_vector_type(2))) float v2f;
typedef __attribute__((ext_vector_type(8))) float v8f;

// Number of upper-triangle elements in rows < i (row-major triu order):
// off(i) = i*IN_DIM - i*(i-1)/2
__device__ __forceinline__ long long triu_off(int i) {
    return (long long)i * IN_DIM - ((long long)i * (i - 1)) / 2;
}

// ---------------------------------------------------------------------------
// Kernel 1: out[b,o] = bias[o] + sum_c xf[b,c] * W[o,c]   (linear part, 4 MB)
// ---------------------------------------------------------------------------
__global__ __launch_bounds__(256) void qam_linear_kernel(
    const float* __restrict__ xf, const float* __restrict__ W,
    const float* __restrict__ bias, float* __restrict__ out)
{
    const int o = blockIdx.x * 256 + threadIdx.x;   // 0..1023
    const int b = blockIdx.y;                       // 0..63
    const float* __restrict__ wr = W + (size_t)o * WSTRIDE;
    const float* __restrict__ xr = xf + b * IN_DIM;
    float acc = bias[o];
#pragma unroll 4
    for (int c = 0; c < IN_DIM; c += 4) {
        float4 w4 = *(const float4*)(wr + c);
        float4 x4 = *(const float4*)(xr + c);
        acc += w4.x * x4.x + w4.y * x4.y + w4.z * x4.z + w4.w * x4.w;
    }
    out[(size_t)b * OUT_DIM + o] = acc;
}

// ---------------------------------------------------------------------------
// Kernel 2: quadratic part via V_WMMA_F32_16X16X4_F32, split-K over triangle
// rows, accumulated into out with hardware f32 global atomics.
//
// Grid: x = 128 row-groups (group g owns rows i = g + 128*t, t=0..7)
//       y = 4 N-tiles of 256 outputs
// Block: 512 threads = 16 wave32 (4 waves/SIMD); wave w owns N columns
//        [ytile*256 + w*16, +16) and all 64 batches (4 accumulator tiles).
// ---------------------------------------------------------------------------
__global__ __launch_bounds__(512) void qam_quad_kernel(
    const float* __restrict__ xf, const float* __restrict__ W,
    float* __restrict__ out)
{
    extern __shared__ float lds_xf[];   // [BATCH][XS] = 257 KB (padded)

    const int tid = threadIdx.x;

    // Stage activations into LDS (coalesced b128 reads, conflict-free stores).
    for (int idx = tid * 4; idx < BATCH * IN_DIM; idx += 512 * 4) {
        const int b = idx >> 10;          // /IN_DIM
        const int c = idx & (IN_DIM - 1);
        float4 v = *(const float4*)(xf + idx);
        *(float4*)(lds_xf + (size_t)b * XS + c) = v;
    }
    __syncthreads();

    const int g      = blockIdx.x;       // row group 0..127
    const int ntile  = blockIdx.y;       // 0..3
    const int wave   = tid >> 5;         // 0..15
    const int lane   = tid & 31;
    const int lm     = lane & 15;        // M/N index within fragment
    const int khalf  = lane >> 4;        // 0 or 1
    const int klo    = khalf * 2;        // this lane's K sub-offset {0,2}
    const int n_base = ntile * 256 + wave * 16;
    const int o_lane = n_base + lm;      // this lane's W row (output index)

    const float* __restrict__ wrow = W + (size_t)o_lane * WSTRIDE;

    // Per-M-tile LDS row pointers (batch rows 16*mt + lm), offset by klo.
    const float* xrow[4];
#pragma unroll
    for (int mt = 0; mt < 4; ++mt)
        xrow[mt] = lds_xf + (size_t)(16 * mt + lm) * XS + klo;

    v8f acc[4];
#pragma unroll
    for (int mt = 0; mt < 4; ++mt)
        acc[mt] = (v8f){0.f, 0.f, 0.f, 0.f, 0.f, 0.f, 0.f, 0.f};

    for (int t = 0; t < 8; ++t) {
        const int i = g + 128 * t;                    // triangle row
        // W column for pair (i,j) is 1024 + off(i) + (j - i) = col0 + j
        const long long col0 = (long long)IN_DIM + triu_off(i) - (long long)i;
        const float* __restrict__ wp = wrow + col0 + klo;

        float xi[4];
#pragma unroll
        for (int mt = 0; mt < 4; ++mt)
            xi[mt] = lds_xf[(size_t)(16 * mt + lm) * XS + i];

        int j = i;
        // Steady state: blocks of 8 K-steps with a W-stream prefetch ~4KB
        // ahead (global_prefetch_b8; OOB prefetch is speculative & dropped).
        for (; j + 31 < IN_DIM; j += 32) {
            __builtin_prefetch(wp + j + 1024, 0, 1);
#pragma unroll
            for (int u = 0; u < 8; ++u) {
                const int jj = j + 4 * u;
                const v2f bfrag = *(const v2f*)(wp + jj);  // W[o, col0+jj+klo..+1]
#pragma unroll
                for (int mt = 0; mt < 4; ++mt) {
                    v2f afrag;
                    afrag.x = xi[mt] * xrow[mt][jj];       // xf[b,i]*xf[b,jj+klo]
                    afrag.y = xi[mt] * xrow[mt][jj + 1];
                    acc[mt] = __builtin_amdgcn_wmma_f32_16x16x4_f32(
                        false, afrag, false, bfrag, (short)0, acc[mt], false, false);
                }
            }
        }
        // Remaining full K=4 steps.
        for (; j + 3 < IN_DIM; j += 4) {
            const v2f bfrag = *(const v2f*)(wp + j);
#pragma unroll
            for (int mt = 0; mt < 4; ++mt) {
                v2f afrag;
                afrag.x = xi[mt] * xrow[mt][j];
                afrag.y = xi[mt] * xrow[mt][j + 1];
                acc[mt] = __builtin_amdgcn_wmma_f32_16x16x4_f32(
                    false, afrag, false, bfrag, (short)0, acc[mt], false, false);
            }
        }
        // Tail: 1..3 valid columns; zero-mask A and B, clamp W address in-row.
        if (j < IN_DIM) {
            const int c1 = j + klo, c2 = j + klo + 1;
            const float m1 = (c1 < IN_DIM) ? 1.0f : 0.0f;
            const float m2 = (c2 < IN_DIM) ? 1.0f : 0.0f;
            const int cc1 = (c1 < IN_DIM) ? c1 : (IN_DIM - 1);
            const int cc2 = (c2 < IN_DIM) ? c2 : (IN_DIM - 1);
            v2f bfrag;
            bfrag.x = m1 * wrow[col0 + cc1];
            bfrag.y = m2 * wrow[col0 + cc2];
#pragma unroll
            for (int mt = 0; mt < 4; ++mt) {
                const float* xr0 = lds_xf + (size_t)(16 * mt + lm) * XS;
                v2f afrag;
                afrag.x = m1 * xi[mt] * xr0[cc1];
                afrag.y = m2 * xi[mt] * xr0[cc2];
                acc[mt] = __builtin_amdgcn_wmma_f32_16x16x4_f32(
                    false, afrag, false, bfrag, (short)0, acc[mt], false, false);
            }
        }
    }

    // Accumulate split-K partials: C layout (16x16 f32): lanes 0-15 hold M=r,
    // lanes 16-31 hold M=8+r, N = lane%16.
#pragma unroll
    for (int mt = 0; mt < 4; ++mt) {
#pragma unroll
        for (int r = 0; r < 8; ++r) {
            const int b = 16 * mt + 8 * khalf + r;
            unsafeAtomicAdd(out + (size_t)b * OUT_DIM + o_lane, acc[mt][r]);
        }
    }
}

// ---------------------------------------------------------------------------
extern "C" void kernel_launch(void* const* d_in, const int* in_sizes, int n_in,
                              void* d_out, int out_size, void* d_ws, size_t ws_size,
                              hipStream_t stream) {
    (void)in_sizes; (void)n_in; (void)out_size; (void)d_ws; (void)ws_size;
    const float* xf   = (const float*)d_in[0];   // (64, 32, 32) == (64, 1024)
    const float* W    = (const float*)d_in[1];   // (1024, 525824)
    const float* bias = (const float*)d_in[2];   // (1024,)
    float* out = (float*)d_out;                  // (64, 1024)

    // 1) out = bias + xf @ W[:, :1024].T
    qam_linear_kernel<<<dim3(OUT_DIM / 256, BATCH), 256, 0, stream>>>(xf, W, bias, out);

    // 2) out += quadratic-feature GEMM (split-K over triangle rows, atomics)
    const size_t lds_bytes = (size_t)BATCH * XS * sizeof(float); // ~257 KB
    qam_quad_kernel<<<dim3(128, 4), 512, lds_bytes, stream>>>(xf, W, out);
}